// Simple_BridgeNet_FF_37855841747270
// MI455X (gfx1250) — compile-verified
//
#include <hip/hip_runtime.h>
#include <math.h>

#define EPS 1e-5f

typedef float v2f __attribute__((ext_vector_type(2)));
typedef float v8f __attribute__((ext_vector_type(8)));

__device__ __forceinline__ float wave_max32(float v) {
#pragma unroll
  for (int m = 16; m >= 1; m >>= 1) v = fmaxf(v, __shfl_xor(v, m, 32));
  return v;
}
__device__ __forceinline__ float wave_sum32(float v) {
#pragma unroll
  for (int m = 16; m >= 1; m >>= 1) v += __shfl_xor(v, m, 32);
  return v;
}

// Fused PointNet block, one wave32 per point.
// LDS layout (floats):
//   [0,4096)      Wfrag   : W_gcm pre-swizzled into WMMA B fragments [ks][nt][lane*2+j]
//   [4096,4864)   Wposl   : W_pos rows, stride 12
//   [4864,4928)   s1l     : folded BN1 scale
//   [4928,4992)   t1l     : folded BN1 shift (includes b_pos)
//   [4992,5024)   Wattl   : attention weights (K=32)
//   [5024,9376)   Woutl   : W_out rows, stride 68
//   [9376+w*2240) per-wave: hbuf 32x68 (reused for h2), then resb[64]
__global__ __launch_bounds__(256)
void bridgenet_fused(const float* __restrict__ points,
                     const float* __restrict__ features,
                     const int*   __restrict__ gidx,
                     const float* __restrict__ W_pos,
                     const float* __restrict__ b_pos,
                     const float* __restrict__ bn1_g, const float* __restrict__ bn1_b,
                     const float* __restrict__ bn1_m, const float* __restrict__ bn1_v,
                     const float* __restrict__ W_gcm,
                     const float* __restrict__ b_gcm,
                     const float* __restrict__ bn2_g, const float* __restrict__ bn2_b,
                     const float* __restrict__ bn2_m, const float* __restrict__ bn2_v,
                     const float* __restrict__ W_att,
                     const float* __restrict__ b_att_p,
                     const float* __restrict__ W_out,
                     const float* __restrict__ b_out,
                     const float* __restrict__ ln_g, const float* __restrict__ ln_b,
                     float* __restrict__ out,
                     int N, int TOT)
{
  extern __shared__ float smem[];
  float* Wfrag = smem;
  float* Wposl = smem + 4096;
  float* s1l   = smem + 4864;
  float* t1l   = smem + 4928;
  float* Wattl = smem + 4992;
  float* Woutl = smem + 5024;

  const int tid  = threadIdx.x;
  const int lane = tid & 31;
  const int wv   = tid >> 5;
  float* hbuf = smem + 9376 + wv * 2240;  // 32 rows x 68 stride
  float* resb = hbuf + 2176;              // 64 floats

  // ---------------- stage weights into LDS (once per block) ----------------
  for (int f = tid; f < 4096; f += 256) {
    // B-fragment layout for V_WMMA_F32_16X16X4_F32:
    // element (kk, n) of 4x16 tile: lane = n + 16*(kk>=2), vgpr j = kk&1
    int ks = f >> 8;
    int rem = f & 255;
    int nt = rem >> 6;
    int l2 = rem & 63;
    int l = l2 >> 1, j = l2 & 1;
    int c = ks * 4 + 2 * (l >> 4) + j;   // reduction index
    int d = nt * 16 + (l & 15);          // output channel
    Wfrag[f] = W_gcm[d * 64 + c];        // B[c][d] = W_gcm[d][c]
  }
  for (int f = tid; f < 4096; f += 256) {
    int o = f >> 6, c = f & 63;
    Woutl[o * 68 + c] = W_out[f];
  }
  if (tid < 64) {
#pragma unroll
    for (int g = 0; g < 10; ++g) Wposl[tid * 12 + g] = W_pos[tid * 10 + g];
    float s = bn1_g[tid] * rsqrtf(bn1_v[tid] + EPS);
    s1l[tid] = s;
    t1l[tid] = s * (b_pos[tid] - bn1_m[tid]) + bn1_b[tid];
  }
  if (tid < 32) Wattl[tid] = W_att[tid];
  __syncthreads();

  // ---------------- per-lane loop invariants ----------------
  const int half = lane >> 4, lm = lane & 15;
  float s2r[4], t2r[4];
#pragma unroll
  for (int nt = 0; nt < 4; ++nt) {
    int d = nt * 16 + lm;
    float s = bn2_g[d] * rsqrtf(bn2_v[d] + EPS);
    s2r[nt] = s;
    t2r[nt] = s * (b_gcm[d] - bn2_m[d]) + bn2_b[d];
  }
  const int o0 = lane, o1 = lane + 32;
  const float bo0 = b_out[o0], bo1 = b_out[o1];
  const float lg0 = ln_g[o0], lb0 = ln_b[o0];
  const float lg1 = ln_g[o1], lb1 = ln_b[o1];
  const float batt = b_att_p[0];

  const int gw = (blockIdx.x << 3) + wv;   // global wave id
  const int nw = gridDim.x << 3;           // total waves
  const int iters = (TOT + nw - 1) / nw;
  int p = gw;

  for (int it = 0; it < iters; ++it, p += nw) {
    const bool active = (p < TOT);

    // ================ phase 1: build h[k][c] (lane = k) ================
    if (active) {
      if (p + nw < TOT)
        __builtin_prefetch(&gidx[(size_t)(p + nw) * 32 + lane], 0, 1);
      const int b  = p / N;
      const int bN = b * N;
      const float* xip = points + (size_t)p * 3;
      const float xi0 = xip[0], xi1 = xip[1], xi2 = xip[2];
      const int idx = gidx[(size_t)p * 32 + lane];
      const float* pj = points + (size_t)(bN + idx) * 3;
      const float q0 = pj[0], q1 = pj[1], q2 = pj[2];
      const float dx = xi0 - q0, dy = xi1 - q1, dz = xi2 - q2;
      const float dist = sqrtf(dx * dx + dy * dy + dz * dz);
      const float geo[10] = {xi0, xi1, xi2, q0, q1, q2, dx, dy, dz, dist};
      const float4* gf = (const float4*)(features + (size_t)(bN + idx) * 64);
      float* hrow = hbuf + lane * 68;
#pragma unroll 4
      for (int c4 = 0; c4 < 16; ++c4) {
        float4 f4 = gf[c4];
        float4 h4;
#pragma unroll
        for (int j = 0; j < 4; ++j) {
          const int c = c4 * 4 + j;
          const float* wr = Wposl + c * 12;
          float acc = 0.f;
#pragma unroll
          for (int g = 0; g < 10; ++g) acc = fmaf(geo[g], wr[g], acc);
          float v = fmaxf(0.f, fmaf(s1l[c], acc, t1l[c]));
          (&h4.x)[j] = (&f4.x)[j] + v;
        }
        *(float4*)(hrow + c4 * 4) = h4;
      }
    }
    __syncthreads();

    // ====== phase 2: h2 = relu(bn2(h @ Wgcm^T)) via v_wmma_f32_16x16x4_f32 ======
    if (active) {
      v8f z = {0.f, 0.f, 0.f, 0.f, 0.f, 0.f, 0.f, 0.f};
      v8f acc[2][4];
#pragma unroll
      for (int mt = 0; mt < 2; ++mt)
#pragma unroll
        for (int nt = 0; nt < 4; ++nt) acc[mt][nt] = z;

      // A fragment (m,kk): lane = m + 16*(kk>=2), vgpr = kk&1 -> one ds_load_b64
      const float* a0p = hbuf + (0 + lm) * 68 + 2 * half;
      const float* a1p = hbuf + (16 + lm) * 68 + 2 * half;
      const float* bbp = Wfrag + lane * 2;
#pragma unroll 4
      for (int ks = 0; ks < 16; ++ks) {
        v2f a0 = *(const v2f*)(a0p + ks * 4);
        v2f a1 = *(const v2f*)(a1p + ks * 4);
#pragma unroll
        for (int nt = 0; nt < 4; ++nt) {
          v2f bf = *(const v2f*)(bbp + ks * 256 + nt * 64);
          acc[0][nt] = __builtin_amdgcn_wmma_f32_16x16x4_f32(
              false, a0, false, bf, (short)0, acc[0][nt], false, false);
          acc[1][nt] = __builtin_amdgcn_wmma_f32_16x16x4_f32(
              false, a1, false, bf, (short)0, acc[1][nt], false, false);
        }
      }
      // epilogue: folded BN2 + ReLU, write h2 back over hbuf
#pragma unroll
      for (int mt = 0; mt < 2; ++mt)
#pragma unroll
        for (int nt = 0; nt < 4; ++nt) {
#pragma unroll
          for (int r = 0; r < 8; ++r) {
            const int krow = mt * 16 + r + 8 * half;
            float v = fmaxf(0.f, fmaf(s2r[nt], acc[mt][nt][r], t2r[nt]));
            hbuf[krow * 68 + nt * 16 + lm] = v;
          }
        }
    }
    __syncthreads();

    // ====== phase 3a: attention score, softmax over C, weighted max-pool ======
    if (active) {
      float sc0 = 0.f, sc1 = 0.f, m0 = -1e30f, m1 = -1e30f;
#pragma unroll 8
      for (int k = 0; k < 32; ++k) {
        const float wa = Wattl[k];
        const float v0 = hbuf[k * 68 + lane];
        const float v1 = hbuf[k * 68 + lane + 32];
        sc0 = fmaf(v0, wa, sc0);
        sc1 = fmaf(v1, wa, sc1);
        m0 = fmaxf(m0, v0);
        m1 = fmaxf(m1, v1);
      }
      sc0 += batt;
      sc1 += batt;
      const float mx = wave_max32(fmaxf(sc0, sc1));
      const float e0 = __expf(sc0 - mx), e1 = __expf(sc1 - mx);
      const float inv = 1.0f / wave_sum32(e0 + e1);
      const float* fr = features + (size_t)p * 64;
      // max_k(h2 * prob) == prob * max_k(h2) since prob >= 0
      resb[lane]      = e0 * inv * m0 + fr[lane];
      resb[lane + 32] = e1 * inv * m1 + fr[lane + 32];
    }
    __syncthreads();

    // ====== phase 3b: output projection + LayerNorm + ReLU ======
    if (active) {
      float y0 = bo0, y1 = bo1;
      const float4* r4 = (const float4*)resb;
      const float* w0 = Woutl + o0 * 68;
      const float* w1 = Woutl + o1 * 68;
#pragma unroll 4
      for (int c4 = 0; c4 < 16; ++c4) {
        const float4 r = r4[c4];
        const float4 a = *(const float4*)(w0 + c4 * 4);
        const float4 c = *(const float4*)(w1 + c4 * 4);
        y0 = fmaf(r.x, a.x, fmaf(r.y, a.y, fmaf(r.z, a.z, fmaf(r.w, a.w, y0))));
        y1 = fmaf(r.x, c.x, fmaf(r.y, c.y, fmaf(r.z, c.z, fmaf(r.w, c.w, y1))));
      }
      const float s = wave_sum32(y0 + y1);
      const float q = wave_sum32(y0 * y0 + y1 * y1);
      const float mu = s * (1.0f / 64.0f);
      const float var = q * (1.0f / 64.0f) - mu * mu;
      const float rs = rsqrtf(var + EPS);
      float* op = out + (size_t)p * 64;
      op[o0] = fmaxf(0.f, fmaf(lg0, (y0 - mu) * rs, lb0));
      op[o1] = fmaxf(0.f, fmaf(lg1, (y1 - mu) * rs, lb1));
    }
  }
}

extern "C" void kernel_launch(void* const* d_in, const int* in_sizes, int n_in,
                              void* d_out, int out_size, void* d_ws, size_t ws_size,
                              hipStream_t stream) {
  (void)n_in; (void)out_size; (void)d_ws; (void)ws_size;
  const float* points   = (const float*)d_in[0];
  const float* features = (const float*)d_in[1];
  const int*   gidx     = (const int*)  d_in[2];
  const float* W_pos    = (const float*)d_in[3];
  const float* b_pos    = (const float*)d_in[4];
  const float* bn1_g    = (const float*)d_in[5];
  const float* bn1_b    = (const float*)d_in[6];
  const float* bn1_m    = (const float*)d_in[7];
  const float* bn1_v    = (const float*)d_in[8];
  const float* W_gcm    = (const float*)d_in[9];
  const float* b_gcm    = (const float*)d_in[10];
  const float* bn2_g    = (const float*)d_in[11];
  const float* bn2_b    = (const float*)d_in[12];
  const float* bn2_m    = (const float*)d_in[13];
  const float* bn2_v    = (const float*)d_in[14];
  const float* W_att    = (const float*)d_in[15];
  const float* b_att    = (const float*)d_in[16];
  const float* W_out    = (const float*)d_in[17];
  const float* b_out    = (const float*)d_in[18];
  const float* ln_g     = (const float*)d_in[19];
  const float* ln_b     = (const float*)d_in[20];
  float* out = (float*)d_out;

  const int B = 2;
  const int TOT = in_sizes[0] / 3;   // B*N points
  const int N = TOT / B;

  const dim3 block(256);
  const dim3 grid(512);
  const size_t shmem = (size_t)(9376 + 8 * 2240) * sizeof(float);  // ~107 KB
  bridgenet_fused<<<grid, block, shmem, stream>>>(
      points, features, gidx, W_pos, b_pos, bn1_g, bn1_b, bn1_m, bn1_v,
      W_gcm, b_gcm, bn2_g, bn2_b, bn2_m, bn2_v, W_att, b_att,
      W_out, b_out, ln_g, ln_b, out, N, TOT);
}